// SNN_1073741824267
// MI455X (gfx1250) — compile-verified
//
#include <hip/hip_runtime.h>

// SNN forward: cur1 = x@W1^T+b1 (fp32 WMMA), then 100-step LIF recurrence.
// Phase 1 uses V_WMMA_F32_16X16X4_F32 (exact fp32, matches reference math).
// Phase 2 keeps all state in registers, one wave32 per batch row.

typedef float v2f __attribute__((ext_vector_type(2)));
typedef float v8f __attribute__((ext_vector_type(8)));

#define NI 343
#define NH 300
#define NO 5

// ---------------------------------------------------------------------------
// Kernel 1: cur1[B][NH] = x[B][NI] @ W1[NH][NI]^T + b1[NH]
// Block = 256 threads = 8 waves. Block tile: M=32 rows, N=320 (padded).
// Wave (mgrp,ngrp): 16 rows x 80 cols -> 5 accumulators of v8f.
// Main K loop (k=0..336) is branch-free: all lanes in-bounds. One masked
// tail iteration (k=340) uses clamped addresses + value selects (cndmask),
// never exec-mask branches.
// ---------------------------------------------------------------------------
__global__ __launch_bounds__(256) void snn_fc1_wmma(
    const float* __restrict__ X, const float* __restrict__ W1,
    const float* __restrict__ b1, float* __restrict__ cur1)
{
  const int tid  = threadIdx.x;
  const int wave = tid >> 5;
  const int lane = tid & 31;
  const int lo   = lane & 15;   // M index (A) / N index (B,C)
  const int hi   = lane >> 4;   // K-pair half select

  const int mgrp    = wave >> 2;                  // 0..1
  const int ngrp    = wave & 3;                   // 0..3
  const int rowBase = blockIdx.x * 32 + mgrp * 16;
  const int nBase   = ngrp * 80;                  // 5 n-tiles of 16

  v8f acc[5];
  v8f zero = {};
  #pragma unroll
  for (int t = 0; t < 5; ++t) acc[t] = zero;

  // Per-lane A row pointer (always valid rows).
  const float* __restrict__ xp = X + (long)(rowBase + lo) * NI;

  // Per-lane clamped W1 row pointers + validity masks (loop-invariant).
  const float* w1p[5];
  bool nok[5];
  #pragma unroll
  for (int t = 0; t < 5; ++t) {
    const int n = nBase + t * 16 + lo;
    nok[t] = (n < NH);
    const int nc = nok[t] ? n : (NH - 1);         // clamped, always in-bounds
    w1p[t] = W1 + (long)nc * NI;
  }

  // ---- main K loop: k = 0..336, kk+1 <= 339 < 343, fully in-bounds -------
  #pragma unroll 2
  for (int k = 0; k < 340; k += 4) {
    const int kk = k + 2 * hi;
    v2f a;
    a.x = xp[kk];
    a.y = xp[kk + 1];
    #pragma unroll
    for (int t = 0; t < 5; ++t) {
      const float bx = w1p[t][kk];                // unconditional loads
      const float by = w1p[t][kk + 1];
      v2f b;
      b.x = nok[t] ? bx : 0.0f;                   // v_cndmask, no branch
      b.y = nok[t] ? by : 0.0f;
      acc[t] = __builtin_amdgcn_wmma_f32_16x16x4_f32(
                   false, a, false, b, (short)0, acc[t], false, false);
    }
  }

  // ---- tail iteration k = 340: kk in {340,342}; element 343 is padding ---
  {
    const int kk   = 340 + 2 * hi;                // 340 or 342 (in-bounds)
    const bool k1ok = (kk + 1 < NI);              // false only for hi==1
    const int kk1c = k1ok ? (kk + 1) : (NI - 1);  // clamped address
    v2f a;
    a.x = xp[kk];
    {
      const float ay = xp[kk1c];
      a.y = k1ok ? ay : 0.0f;
    }
    #pragma unroll
    for (int t = 0; t < 5; ++t) {
      const float bx = w1p[t][kk];
      const float by = w1p[t][kk1c];
      v2f b;
      b.x = nok[t]          ? bx : 0.0f;
      b.y = (nok[t] & k1ok) ? by : 0.0f;
      acc[t] = __builtin_amdgcn_wmma_f32_16x16x4_f32(
                   false, a, false, b, (short)0, acc[t], false, false);
    }
  }

  // C/D layout: VGPR r holds M = r + 8*hi, N = lo. Add bias and store.
  #pragma unroll
  for (int t = 0; t < 5; ++t) {
    const int n = nBase + t * 16 + lo;
    if (n < NH) {
      const float bias = b1[n];
      #pragma unroll
      for (int r = 0; r < 8; ++r) {
        const int row = rowBase + hi * 8 + r;
        cur1[(long)row * NH + n] = acc[t][r] + bias;
      }
    }
  }
}

// ---------------------------------------------------------------------------
// Kernel 2: 100-step LIF recurrence. One wave per batch row.
// mem1[300] striped 10-per-lane in registers; W2 held in registers;
// spk1 @ W2^T = 50 FMAs/lane + 5-stage wave32 butterfly per output.
// ---------------------------------------------------------------------------
__global__ __launch_bounds__(256) void snn_steps(
    const float* __restrict__ cur1, const float* __restrict__ W2,
    const float* __restrict__ b2, const float* __restrict__ beta1,
    const float* __restrict__ beta2, float* __restrict__ out,
    int B, int T)
{
  const int wave = threadIdx.x >> 5;
  const int lane = threadIdx.x & 31;
  const int row  = blockIdx.x * 8 + wave;
  if (row >= B) return;   // B % 8 == 0, uniform per block in practice

  float b1c = *beta1;
  b1c = b1c < 0.0f ? 0.0f : (b1c > 1.0f ? 1.0f : b1c);

  float b2c[NO], bb2[NO];
  #pragma unroll
  for (int o = 0; o < NO; ++o) {
    float v = beta2[o];
    b2c[o] = v < 0.0f ? 0.0f : (v > 1.0f ? 1.0f : v);
    bb2[o] = b2[o];
  }

  // hidden state: lane owns h = lane + 32*j, j = 0..9 (h < 300 valid)
  float c[10], m1[10], w[NO][10];
  #pragma unroll
  for (int j = 0; j < 10; ++j) {
    const int h = lane + 32 * j;
    const bool ok = h < NH;
    const int hc = ok ? h : (NH - 1);
    const float cv = cur1[(long)row * NH + hc];
    c[j]  = ok ? cv : 0.0f;
    m1[j] = 0.0f;
    #pragma unroll
    for (int o = 0; o < NO; ++o) {
      const float wv = W2[o * NH + hc];
      w[o][j] = ok ? wv : 0.0f;   // invalid h: c=0 => never spikes anyway
    }
  }

  float m2[NO];
  #pragma unroll
  for (int o = 0; o < NO; ++o) m2[o] = 0.0f;

  float* __restrict__ spkOut = out;
  float* __restrict__ memOut = out + (long)T * B * NO;

  for (int s = 0; s < T; ++s) {
    float acc[NO];
    #pragma unroll
    for (int o = 0; o < NO; ++o) acc[o] = 0.0f;

    // layer 1: reset-subtract LIF update + partial spk1 . W2^T
    #pragma unroll
    for (int j = 0; j < 10; ++j) {
      float m   = m1[j];
      float rst = (m > 1.0f) ? 1.0f : 0.0f;      // reset from incoming mem
      m = fmaf(b1c, m, c[j]) - rst;              // THRESHOLD = 1.0
      m1[j] = m;
      float spk = (m > 1.0f) ? 1.0f : 0.0f;
      #pragma unroll
      for (int o = 0; o < NO; ++o) acc[o] = fmaf(spk, w[o][j], acc[o]);
    }

    // wave32 butterfly reduction over h
    #pragma unroll
    for (int o = 0; o < NO; ++o) {
      float a = acc[o];
      a += __shfl_xor(a, 16, 32);
      a += __shfl_xor(a,  8, 32);
      a += __shfl_xor(a,  4, 32);
      a += __shfl_xor(a,  2, 32);
      a += __shfl_xor(a,  1, 32);
      acc[o] = a;
    }

    // layer 2 LIF (all lanes redundantly, avoids divergence in state)
    float spk2[NO];
    #pragma unroll
    for (int o = 0; o < NO; ++o) {
      float cur2 = acc[o] + bb2[o];
      float m    = m2[o];
      float rst  = (m > 1.0f) ? 1.0f : 0.0f;
      m = fmaf(b2c[o], m, cur2) - rst;
      m2[o]  = m;
      spk2[o] = (m > 1.0f) ? 1.0f : 0.0f;
    }

    // lanes 0..4 write spk_rec / mem_rec for this (step,row)
    float vs = spk2[0], vm = m2[0];
    #pragma unroll
    for (int o = 1; o < NO; ++o) {
      vs = (lane == o) ? spk2[o] : vs;
      vm = (lane == o) ? m2[o]   : vm;
    }
    const long off = (long)s * B * NO + (long)row * NO + lane;
    if (lane < NO) {
      spkOut[off] = vs;
      memOut[off] = vm;
    }
  }
}

// ---------------------------------------------------------------------------
extern "C" void kernel_launch(void* const* d_in, const int* in_sizes, int n_in,
                              void* d_out, int out_size, void* d_ws, size_t ws_size,
                              hipStream_t stream) {
  const float* X     = (const float*)d_in[0];  // [B, NI]
  const float* W1    = (const float*)d_in[1];  // [NH, NI]
  const float* b1    = (const float*)d_in[2];  // [NH]
  const float* W2    = (const float*)d_in[3];  // [NO, NH]
  const float* b2    = (const float*)d_in[4];  // [NO]
  const float* beta1 = (const float*)d_in[5];  // scalar
  const float* beta2 = (const float*)d_in[6];  // [NO]
  // d_in[7] = num_steps (device int); T recovered from out_size instead.

  const int B = in_sizes[0] / NI;              // 16384
  const int T = out_size / (2 * B * NO);       // 100

  float* cur1 = (float*)d_ws;                  // [B, NH] fp32 = 19.7 MB

  snn_fc1_wmma<<<B / 32, 256, 0, stream>>>(X, W1, b1, cur1);
  snn_steps<<<(B + 7) / 8, 256, 0, stream>>>(cur1, W2, b2, beta1, beta2,
                                             (float*)d_out, B, T);
}